// my_GCN_2087354105940
// MI455X (gfx1250) — compile-verified
//
#include <hip/hip_runtime.h>
#include <hip/hip_bf16.h>

typedef __attribute__((ext_vector_type(2))) float v2f;
typedef __attribute__((ext_vector_type(8))) float v8f;

#define IN_F 128
#define H_F  64

// ---------------------------------------------------------------- utilities
__global__ void gcn_zero_f32(float* __restrict__ p, long long n) {
    long long i = (long long)blockIdx.x * blockDim.x + threadIdx.x;
    long long stride = (long long)gridDim.x * blockDim.x;
    for (; i < n; i += stride) p[i] = 0.0f;
}

// deg_out[src[e]] += 1 ; deg_in[dst[e]] += 1   (hardware f32 atomics)
__global__ void gcn_degrees(const int* __restrict__ src, const int* __restrict__ dst,
                            float* __restrict__ deg_out, float* __restrict__ deg_in, int E) {
    int i = blockIdx.x * blockDim.x + threadIdx.x;
    int stride = gridDim.x * blockDim.x;
    for (; i < E; i += stride) {
        unsafeAtomicAdd(&deg_out[src[i]], 1.0f);
        unsafeAtomicAdd(&deg_in[dst[i]], 1.0f);
    }
}

// in-place deg -> (deg>0 ? rsqrt(deg) : 0); operates on [deg_out|deg_in] contiguously
__global__ void gcn_norms(float* __restrict__ deg, long long n) {
    long long i = (long long)blockIdx.x * blockDim.x + threadIdx.x;
    if (i >= n) return;
    float d = deg[i];
    deg[i] = (d > 0.0f) ? __frsqrt_rn(d) : 0.0f;
}

// ---------------------------------------------------------------- WMMA GEMM
// H[n, 0:NCOLS] = (X[n,0:128] * norm[n]) @ W[128, NCOLS]
// Block: NCOLS/16 waves (wave w -> 16-col tile), 16 rows per block.
// Uses V_WMMA_F32_16X16X4_F32 (full f32 precision, D = A(16x4) x B(4x16) + C).
template <int NCOLS>
__global__ void gcn_gemm_wmma(const float* __restrict__ X, const float* __restrict__ norm,
                              const float* __restrict__ W, float* __restrict__ H, int N) {
    __shared__ float lds[16 * 128];           // A tile: 16 rows x K=128, norm fused
    const int tile0 = blockIdx.x * 16;

    // cooperative staging of the A tile (norm_out applied here)
    for (int i = threadIdx.x; i < 16 * 128; i += blockDim.x) {
        int r = i >> 7;            // 0..15
        int k = i & 127;
        int row = tile0 + r;
        lds[i] = (row < N) ? X[(size_t)row * 128 + k] * norm[row] : 0.0f;
    }
    __syncthreads();

    const int lane = threadIdx.x & 31;
    const int wv   = threadIdx.x >> 5;
    const int cb   = wv * 16;                 // column-tile base
    const int m    = lane & 15;               // row (A) / col (B,C) within tile
    const int kh   = (lane >> 4) << 1;        // lanes 16..31 hold K+2,K+3

    __builtin_prefetch(&W[(size_t)kh * NCOLS + cb + m], 0, 3);

    v8f acc = {};
    for (int kk = 0; kk < 128; kk += 4) {
        v2f a, b;
        // A 16x4 f32 layout (ISA 7.12.2): lane<16 -> K=kk,kk+1 ; lane>=16 -> K=kk+2,kk+3
        a.x = lds[m * 128 + kk + kh];
        a.y = lds[m * 128 + kk + kh + 1];
        // B 4x16 f32: row K striped across lanes; VGPR0={K0|K2}, VGPR1={K1|K3}
        b.x = W[(size_t)(kk + kh) * NCOLS + cb + m];
        b.y = W[(size_t)(kk + kh + 1) * NCOLS + cb + m];
        acc = __builtin_amdgcn_wmma_f32_16x16x4_f32(
            /*neg_a=*/false, a, /*neg_b=*/false, b,
            /*c_mod=*/(short)0, acc, /*reuse_a=*/false, /*reuse_b=*/false);
    }

    // C/D 16x16 f32: VGPR r -> M=r (lanes 0-15) / M=r+8 (lanes 16-31), N = lane&15
    const int row0 = tile0 + ((lane >> 4) << 3);
    for (int r = 0; r < 8; ++r) {
        int row = row0 + r;
        if (row < N) H[(size_t)row * NCOLS + cb + m] = acc[r];
    }
}

// ---------------------------------------------------------------- edge scatter
// One wave per edge; 128 feats -> float4 per lane. agg[dst] += h[src].
__global__ void gcn_scatter128(const float* __restrict__ h, const int* __restrict__ src,
                               const int* __restrict__ dst, float* __restrict__ agg, int E) {
    int e = blockIdx.x * (blockDim.x >> 5) + (threadIdx.x >> 5);
    if (e >= E) return;
    int lane = threadIdx.x & 31;
    int s = src[e], d = dst[e];
    const float4 v = *(const float4*)(h + (size_t)s * 128 + lane * 4);
    float* base = agg + (size_t)d * 128 + lane * 4;
    unsafeAtomicAdd(base + 0, v.x);
    unsafeAtomicAdd(base + 1, v.y);
    unsafeAtomicAdd(base + 2, v.z);
    unsafeAtomicAdd(base + 3, v.w);
}

// One wave per edge; 64 feats -> float2 per lane.
__global__ void gcn_scatter64(const float* __restrict__ h, const int* __restrict__ src,
                              const int* __restrict__ dst, float* __restrict__ agg, int E) {
    int e = blockIdx.x * (blockDim.x >> 5) + (threadIdx.x >> 5);
    if (e >= E) return;
    int lane = threadIdx.x & 31;
    int s = src[e], d = dst[e];
    const float2 v = *(const float2*)(h + (size_t)s * 64 + lane * 2);
    float* base = agg + (size_t)d * 64 + lane * 2;
    unsafeAtomicAdd(base + 0, v.x);
    unsafeAtomicAdd(base + 1, v.y);
}

// ---------------------------------------------------------------- epilogue
// out[n,f] = relu(agg[n,f] * norm_in[n] + b[f])   (out may alias agg)
template <int F>
__global__ void gcn_epilogue(const float* __restrict__ agg, const float* __restrict__ norm_in,
                             const float* __restrict__ b, float* __restrict__ out, int N) {
    long long i = (long long)blockIdx.x * blockDim.x + threadIdx.x;
    if (i >= (long long)N * F) return;
    int n = (int)(i / F);
    int f = (int)(i % F);
    float v = agg[i] * norm_in[n] + b[f];
    out[i] = v > 0.0f ? v : 0.0f;
}

// ---------------------------------------------------------------- launcher
extern "C" void kernel_launch(void* const* d_in, const int* in_sizes, int n_in,
                              void* d_out, int out_size, void* d_ws, size_t ws_size,
                              hipStream_t stream) {
    const float* x  = (const float*)d_in[0];
    const float* W1 = (const float*)d_in[1];
    const float* b1 = (const float*)d_in[2];
    const float* W2 = (const float*)d_in[3];
    const float* b2 = (const float*)d_in[4];
    const int*  src = (const int*)d_in[5];
    const int*  dst = (const int*)d_in[6];
    float* out = (float*)d_out;

    const int N = in_sizes[0] / IN_F;   // 100000
    const int E = in_sizes[5];          // 1600000

    // workspace layout (floats): [deg_out N][deg_in N][agg1 128N][agg2 64N][h1pre 128N][h2pre 64N]
    float* ws       = (float*)d_ws;
    float* deg_out  = ws;                         // becomes norm_out in-place
    float* deg_in   = ws + (size_t)N;             // becomes norm_in in-place
    float* agg1     = ws + (size_t)2 * N;         // becomes h1 (post-relu) in-place
    float* agg2     = agg1 + (size_t)N * IN_F;
    float* h1pre    = agg2 + (size_t)N * H_F;
    float* h2pre    = h1pre + (size_t)N * IN_F;

    // 1) zero [deg_out|deg_in|agg1|agg2] (contiguous region of (2+128+64)*N floats)
    {
        long long nz = (long long)N * (2 + IN_F + H_F);
        gcn_zero_f32<<<2048, 256, 0, stream>>>(ws, nz);
    }

    // 2) degrees, 3) norms (in place over the contiguous [deg_out|deg_in] block)
    gcn_degrees<<<(E + 255) / 256, 256, 0, stream>>>(src, dst, deg_out, deg_in, E);
    gcn_norms<<<(2 * N + 255) / 256, 256, 0, stream>>>(deg_out, (long long)2 * N);

    const int rowTiles = (N + 15) / 16;

    // 4) layer-1 GEMM: h1pre = (x * norm_out) @ W1   (8 waves = 128 cols)
    gcn_gemm_wmma<IN_F><<<rowTiles, 256, 0, stream>>>(x, deg_out, W1, h1pre, N);

    // 5) layer-1 scatter-add over edges (8 edges per 256-thread block)
    gcn_scatter128<<<(E + 7) / 8, 256, 0, stream>>>(h1pre, src, dst, agg1, E);

    // 6) h1 = relu(agg1 * norm_in + b1), in place
    gcn_epilogue<IN_F><<<(int)(((long long)N * IN_F + 255) / 256), 256, 0, stream>>>(
        agg1, deg_in, b1, agg1, N);

    // 7) layer-2 GEMM: h2pre = (h1 * norm_out) @ W2  (4 waves = 64 cols)
    gcn_gemm_wmma<H_F><<<rowTiles, 128, 0, stream>>>(agg1, deg_out, W2, h2pre, N);

    // 8) layer-2 scatter-add
    gcn_scatter64<<<(E + 7) / 8, 256, 0, stream>>>(h2pre, src, dst, agg2, E);

    // 9) out = relu(agg2 * norm_in + b2)
    gcn_epilogue<H_F><<<(int)(((long long)N * H_F + 255) / 256), 256, 0, stream>>>(
        agg2, deg_in, b2, out, N);
}